// ProductLayer_52802327937173
// MI455X (gfx1250) — compile-verified
//
#include <hip/hip_runtime.h>
#include <stdint.h>

// ProductLayer: lp[b,p] = sum_d emb[b,row[p],d]*emb[b,col[p],d]*w[p,d]
// BATCH=4096, NUM_FIELDS=33, EMB_DIM=64, PAIR_NUM=528.
//
// MI455X (gfx1250, wave32) memory-bound fp32 kernel:
//  - no GEMM structure (both product factors depend on both output indices),
//    so WMMA would run at 1/16 diagonal efficiency -> use dual-issue f32 FMA.
//  - embeddings staged into LDS with gfx1250 async global->LDS copies
//    (global_load_async_to_lds_b128), double buffered, 2 samples per buffer,
//    synchronized with s_wait_asynccnt.
//  - per-pair weight row (64 floats) register-cached across all 32 samples a
//    block processes; weights stay L2-resident.

#define NF     33
#define ED     64
#define NPAIR  528
#define PCHUNK 176   // pairs per block == blockDim.x ; 3 * 176 == 528
#define SCHUNK 32    // samples per block
#define SPB    2     // samples per LDS buffer
#define NSTAGE (SCHUNK / SPB)          // 16 stages
#define CHUNKS (SPB * NF * ED * 4 / 16)  // 16B chunks per stage = 1056 = 6*176
#define NISSUE (CHUNKS / PCHUNK)       // async issues per wave per stage = 6

struct f4 { float x, y, z, w; };

__global__ __launch_bounds__(PCHUNK) void ProductLayer_52802327937173_kernel(
    const float* __restrict__ emb,   // [BATCH][NF][ED]
    const float* __restrict__ wgt,   // [NPAIR][ED]
    float* __restrict__ out)         // [BATCH][NPAIR]
{
    __shared__ __align__(16) float smem[2][SPB * NF * ED];   // 2 x 16896 B

    const int t     = threadIdx.x;            // 0..175
    const int sBase = blockIdx.x * SCHUNK;    // sample chunk base
    const int pBase = blockIdx.y * PCHUNK;    // pair chunk base
    const int p     = pBase + t;              // this thread's pair, < 528

    // ---- decode pair index -> (i, j), i < j < 33 (lexicographic combos) ----
    int i = 0, rem = p;
    while (rem >= (NF - 1 - i)) { rem -= (NF - 1 - i); ++i; }
    const int j = i + 1 + rem;

    // ---- register-cache this pair's 64 weights (16 x float4 = 64 VGPRs) ----
    f4 wreg[16];
    {
        const f4* wrow = reinterpret_cast<const f4*>(wgt + (size_t)p * ED);
        #pragma unroll
        for (int k = 0; k < 16; ++k) wreg[k] = wrow[k];
    }

    // LDS byte offsets of the two buffers. Flat shared pointers carry the
    // workgroup-relative LDS byte offset in their low 32 bits (CDNA5 aperture
    // mapping), which is exactly what async-to-LDS VDST expects.
    const unsigned ldsBase0 = (unsigned)(size_t)(&smem[0][0]);
    const unsigned ldsBase1 = (unsigned)(size_t)(&smem[1][0]);

    // ---- async stage of SPB samples (16896 B = 1056 x 16B = 6 x 176 lanes) --
    // Every wave has EXEC!=0 on all NISSUE issues, so each wave's ASYNCcnt
    // advances by exactly NISSUE per stage -> fixed-immediate waits are valid.
    auto stage = [&](int st, int buf) {
        const unsigned long long src =
            (unsigned long long)(emb) +
            (unsigned long long)(sBase + st * SPB) * (NF * ED * 4ull);
        const unsigned lbase = buf ? ldsBase1 : ldsBase0;
        #pragma unroll
        for (int it = 0; it < NISSUE; ++it) {
            const unsigned c  = (unsigned)(it * PCHUNK + t);   // 0..1055
            const unsigned la = lbase + c * 16u;
            const unsigned long long ga = src + (unsigned long long)c * 16ull;
            asm volatile("global_load_async_to_lds_b128 %0, %1, off"
                         :: "v"(la), "v"(ga) : "memory");
        }
    };

    stage(0, 0);

    #pragma unroll 1
    for (int st = 0; st < NSTAGE; ++st) {
        const int buf = st & 1;
        if (st + 1 < NSTAGE) {
            stage(st + 1, buf ^ 1);
            // stage(st) + stage(st+1) in flight (<=12); async loads complete
            // in order, so <=6 outstanding means stage(st) has fully landed.
            asm volatile("s_wait_asynccnt 6" ::: "memory");
        } else {
            asm volatile("s_wait_asynccnt 0" ::: "memory");
        }
        __syncthreads();   // all waves' slices of stage(st) visible in LDS

        #pragma unroll
        for (int ss = 0; ss < SPB; ++ss) {
            const float* base = &smem[buf][ss * NF * ED];
            const f4* prow = reinterpret_cast<const f4*>(base + i * ED);
            const f4* qrow = reinterpret_cast<const f4*>(base + j * ED);

            float ax = 0.f, ay = 0.f, az = 0.f, aw = 0.f;
            #pragma unroll
            for (int k = 0; k < 16; ++k) {
                const f4 a = prow[k];
                const f4 b = qrow[k];
                const f4 c = wreg[k];
                ax = fmaf(a.x * b.x, c.x, ax);
                ay = fmaf(a.y * b.y, c.y, ay);
                az = fmaf(a.z * b.z, c.z, az);
                aw = fmaf(a.w * b.w, c.w, aw);
            }
            const int s = st * SPB + ss;
            out[(size_t)(sBase + s) * NPAIR + p] = (ax + ay) + (az + aw);
        }

        __syncthreads();   // compute done before stage(st+2) overwrites buf
    }
}

extern "C" void kernel_launch(void* const* d_in, const int* in_sizes, int n_in,
                              void* d_out, int out_size, void* d_ws, size_t ws_size,
                              hipStream_t stream) {
    (void)n_in; (void)out_size; (void)d_ws; (void)ws_size;
    const float* emb = (const float*)d_in[0];   // [B,33,64] fp32
    const float* wgt = (const float*)d_in[1];   // [528,64] fp32
    float* out = (float*)d_out;                 // [B,528] fp32

    const int batch = in_sizes[0] / (NF * ED);  // 4096
    dim3 grid(batch / SCHUNK, NPAIR / PCHUNK, 1);   // (128, 3)
    dim3 block(PCHUNK, 1, 1);                        // 176 threads
    ProductLayer_52802327937173_kernel<<<grid, block, 0, stream>>>(emb, wgt, out);
}